// GCN_52209622450204
// MI455X (gfx1250) — compile-verified
//
#include <hip/hip_runtime.h>
#include <hip/hip_bf16.h>

typedef __attribute__((ext_vector_type(2))) float v2f;
typedef __attribute__((ext_vector_type(8))) float v8f;

#define IN_DIM 256
#define HID    64
#define OUTD   10

// ---------------------------------------------------------------------------
// Utility kernels
// ---------------------------------------------------------------------------
__global__ void zero_f32(float* __restrict__ p, long long n) {
    long long i = (long long)blockIdx.x * blockDim.x + threadIdx.x;
    if (i < n) p[i] = 0.0f;
}

// deg[c] += w[e]  (in-degree by weight)
__global__ void deg_kernel(const long long* __restrict__ ei,
                           const float* __restrict__ w,
                           float* __restrict__ deg, int E) {
    int e = blockIdx.x * blockDim.x + threadIdx.x;
    if (e < E) {
        long long c = ei[(long long)E + e];
        atomicAdd(&deg[c], w[e]);
    }
}

// deg -> dinv in place: deg>0 ? rsqrt(max(deg,1e-12)) : 0
__global__ void dinv_kernel(float* __restrict__ deg, int n) {
    int i = blockIdx.x * blockDim.x + threadIdx.x;
    if (i < n) {
        float d = deg[i];
        deg[i] = (d > 0.0f) ? rsqrtf(fmaxf(d, 1e-12f)) : 0.0f;
    }
}

// norm[e] = dinv[row] * w[e] * dinv[col]   (computed once, reused by 3 layers)
__global__ void norm_kernel(const long long* __restrict__ ei,
                            const float* __restrict__ w,
                            const float* __restrict__ dinv,
                            float* __restrict__ norm, int E) {
    int e = blockIdx.x * blockDim.x + threadIdx.x;
    if (e < E) {
        long long r = ei[e];
        long long c = ei[(long long)E + e];
        norm[e] = dinv[r] * w[e] * dinv[c];
    }
}

// ---------------------------------------------------------------------------
// fp32 WMMA GEMM:  H[N, OUT] = X[N, K] * W[OUT, K]^T
//
// Block = 2 M-subtiles x NT N-tiles = 64*NT threads; one wave per 16x16 tile.
// Stage A-tile (32 rows x K, a contiguous chunk of X) and the whole weight
// matrix into LDS with coalesced float4 copies.  LDS row stride = K+4 words:
// the 16 row-strided lanes of a ds_load_b64 then map to banks {4L,4L+1}
// (lanes 0-15, K-half 0) and {4L+2,4L+3} (lanes 16-31, K-half 2) -> no bank
// conflicts across the 64 banks.
//
// A-lane layout (ISA 7.12.2, 32-bit A 16x4): lanes 0-15 -> K=0,1 ; 16-31 -> K=2,3.
// C/D layout: VGPR i -> row m_base + i + (lane>=16 ? 8 : 0), col n_base + lane%16.
// ---------------------------------------------------------------------------
template <int K, int OUT, int NT>
__global__ __launch_bounds__(64 * NT)
void gemm_wmma_f32(const float* __restrict__ X,
                   const float* __restrict__ W,
                   float* __restrict__ H, int Nrows) {
    constexpr int LDA = K + 4;           // padded LDS row stride (words)
    constexpr int KV  = K / 4;           // float4s per row
    __shared__ float sA[32 * LDA];
    __shared__ float sB[OUT * LDA];

    const int tid = threadIdx.x;
    constexpr int NTHREADS = 64 * NT;
    const int m0 = blockIdx.x * 32;

    // ---- cooperative, fully-coalesced stage of A tile ----
#pragma unroll 2
    for (int i = tid; i < 32 * KV; i += NTHREADS) {
        int r  = i / KV;
        int c4 = i % KV;
        float4 v = make_float4(0.f, 0.f, 0.f, 0.f);
        int gr = m0 + r;
        if (gr < Nrows) v = *(const float4*)(X + (long long)gr * K + c4 * 4);
        *(float4*)(&sA[r * LDA + c4 * 4]) = v;
    }
    // ---- stage full weight matrix ----
#pragma unroll 2
    for (int i = tid; i < OUT * KV; i += NTHREADS) {
        int r  = i / KV;
        int c4 = i % KV;
        float4 v = *(const float4*)(W + r * K + c4 * 4);
        *(float4*)(&sB[r * LDA + c4 * 4]) = v;
    }
    __syncthreads();

    const int wave  = tid >> 5;
    const int lane  = tid & 31;
    const int mt    = wave / NT;                 // 0..1 (NT is constexpr)
    const int nt    = wave % NT;                 // 0..NT-1
    const int lhalf = lane & 15;
    const int khalf = (lane >> 4) * 2;           // 0 or 2

    const float* ap = &sA[(mt * 16 + lhalf) * LDA + khalf];
    int brow = nt * 16 + lhalf; if (brow >= OUT) brow = OUT - 1;
    const float* bp = &sB[brow * LDA + khalf];

    v8f acc = {0.f,0.f,0.f,0.f,0.f,0.f,0.f,0.f};
#pragma unroll
    for (int k = 0; k < K; k += 4) {
        v2f a = *(const v2f*)(ap + k);
        v2f b = *(const v2f*)(bp + k);
        acc = __builtin_amdgcn_wmma_f32_16x16x4_f32(
            /*neg_a=*/false, a, /*neg_b=*/false, b,
            /*c_mod=*/(short)0, acc, /*reuse_a=*/false, /*reuse_b=*/false);
    }

    const int col  = nt * 16 + lhalf;
    const int srow = m0 + mt * 16 + (lane >> 4) * 8;
    if (col < OUT) {
#pragma unroll
        for (int i = 0; i < 8; ++i) {
            int r = srow + i;
            if (r < Nrows) H[(long long)r * OUT + col] = acc[i];
        }
    }
}

// ---------------------------------------------------------------------------
// Scatter-add aggregation, dim=64: 16 threads per edge, float4 gather each.
// out[col] += h[row] * norm[e]
// ---------------------------------------------------------------------------
__global__ void scatter64(const float* __restrict__ h,
                          const float* __restrict__ norm,
                          const long long* __restrict__ ei,
                          float* __restrict__ out, int E) {
    long long idx = (long long)blockIdx.x * blockDim.x + threadIdx.x;
    long long total = (long long)E * 16;
    if (idx >= total) return;
    int e  = (int)(idx >> 4);
    int d4 = (int)(idx & 15) * 4;
    long long r = ei[e];
    long long c = ei[(long long)E + e];
    float nv = norm[e];
    const float4 hv = *(const float4*)(h + r * 64 + d4);
    float* op = out + c * 64 + d4;
    atomicAdd(op + 0, hv.x * nv);
    atomicAdd(op + 1, hv.y * nv);
    atomicAdd(op + 2, hv.z * nv);
    atomicAdd(op + 3, hv.w * nv);
}

// Generic small-dim scatter (layer 3, dim=10): one thread per edge.
__global__ void scatter_dim(const float* __restrict__ h,
                            const float* __restrict__ norm,
                            const long long* __restrict__ ei,
                            float* __restrict__ out, int E, int dim) {
    int e = blockIdx.x * blockDim.x + threadIdx.x;
    if (e >= E) return;
    long long r = ei[e];
    long long c = ei[(long long)E + e];
    float nv = norm[e];
    const float* hp = h + r * dim;
    float* op = out + c * dim;
    for (int d = 0; d < dim; ++d) atomicAdd(op + d, hp[d] * nv);
}

// a = relu(a + b[d])
__global__ void bias_relu(float* __restrict__ a, const float* __restrict__ b,
                          long long n, int dim) {
    long long i = (long long)blockIdx.x * blockDim.x + threadIdx.x;
    if (i < n) {
        int d = (int)(i % dim);
        a[i] = fmaxf(a[i] + b[d], 0.0f);
    }
}

__global__ void bias_add(float* __restrict__ a, const float* __restrict__ b,
                         long long n, int dim) {
    long long i = (long long)blockIdx.x * blockDim.x + threadIdx.x;
    if (i < n) {
        int d = (int)(i % dim);
        a[i] += b[d];
    }
}

// ---------------------------------------------------------------------------
// Launch
// ---------------------------------------------------------------------------
static inline size_t align512(size_t x) { return (x + 511) & ~(size_t)511; }

extern "C" void kernel_launch(void* const* d_in, const int* in_sizes, int n_in,
                              void* d_out, int out_size, void* d_ws, size_t ws_size,
                              hipStream_t stream) {
    const float*     x  = (const float*)d_in[0];
    const long long* ei = (const long long*)d_in[1];   // int64 edge_index [2, E]
    const float*     w  = (const float*)d_in[2];
    const float*     W1 = (const float*)d_in[3];
    const float*     b1 = (const float*)d_in[4];
    const float*     W2 = (const float*)d_in[5];
    const float*     b2 = (const float*)d_in[6];
    const float*     W3 = (const float*)d_in[7];
    const float*     b3 = (const float*)d_in[8];
    float* out = (float*)d_out;

    const int N = in_sizes[0] / IN_DIM;   // 100000
    const int E = in_sizes[2];            // 3200000

    // workspace layout
    char* ws = (char*)d_ws;
    size_t off = 0;
    float* deg  = (float*)(ws + off); off += align512((size_t)N * 4);
    float* norm = (float*)(ws + off); off += align512((size_t)E * 4);
    float* h    = (float*)(ws + off); off += align512((size_t)N * HID * 4);
    float* agg  = (float*)(ws + off); off += align512((size_t)N * HID * 4);
    (void)ws_size;

    const int T = 256;
    dim3 blk(T);
    auto blocks = [](long long n, int t) { return (unsigned)((n + t - 1) / t); };

    // ---- normalization coefficients (computed once, reused for all layers) ----
    zero_f32<<<blocks(N, T), blk, 0, stream>>>(deg, N);
    deg_kernel<<<blocks(E, T), blk, 0, stream>>>(ei, w, deg, E);
    dinv_kernel<<<blocks(N, T), blk, 0, stream>>>(deg, N);
    norm_kernel<<<blocks(E, T), blk, 0, stream>>>(ei, w, deg, norm, E);

    const unsigned gemm_grid = blocks(N, 32);

    // ---- layer 1: h = x @ W1^T ; agg = scatter(h) ; agg = relu(agg + b1) ----
    gemm_wmma_f32<IN_DIM, HID, 4><<<gemm_grid, dim3(256), 0, stream>>>(x, W1, h, N);
    zero_f32<<<blocks((long long)N * HID, T), blk, 0, stream>>>(agg, (long long)N * HID);
    scatter64<<<blocks((long long)E * 16, T), blk, 0, stream>>>(h, norm, ei, agg, E);
    bias_relu<<<blocks((long long)N * HID, T), blk, 0, stream>>>(agg, b1, (long long)N * HID, HID);

    // ---- layer 2 ----
    gemm_wmma_f32<HID, HID, 4><<<gemm_grid, dim3(256), 0, stream>>>(agg, W2, h, N);
    zero_f32<<<blocks((long long)N * HID, T), blk, 0, stream>>>(agg, (long long)N * HID);
    scatter64<<<blocks((long long)E * 16, T), blk, 0, stream>>>(h, norm, ei, agg, E);
    bias_relu<<<blocks((long long)N * HID, T), blk, 0, stream>>>(agg, b2, (long long)N * HID, HID);

    // ---- layer 3: logits into d_out ----
    gemm_wmma_f32<HID, OUTD, 1><<<gemm_grid, dim3(64), 0, stream>>>(agg, W3, h, N);
    zero_f32<<<blocks((long long)N * OUTD, T), blk, 0, stream>>>(out, (long long)N * OUTD);
    scatter_dim<<<blocks(E, T), blk, 0, stream>>>(h, norm, ei, out, E, OUTD);
    bias_add<<<blocks((long long)N * OUTD, T), blk, 0, stream>>>(out, b3, (long long)N * OUTD, OUTD);
}